// TripletHardLoss_12086037971165
// MI455X (gfx1250) — compile-verified
//
#include <hip/hip_runtime.h>
#include <stdint.h>

#define N_ROWS 4096
#define DIM    256
#define MARGIN 0.3f

typedef float v2f __attribute__((ext_vector_type(2)));
typedef float v8f __attribute__((ext_vector_type(8)));

// ---------------------------------------------------------------------------
// Kernel 1: sq[i] = ||X_i||^2 ; init ap[i]=0 bits (max identity for d>=0),
// an[i]=+inf bits (min identity). One wave per row, 8 rows per block.
// ---------------------------------------------------------------------------
__global__ void tl_rownorm_init(const float* __restrict__ X,
                                float* __restrict__ sq,
                                unsigned* __restrict__ ap,
                                unsigned* __restrict__ an)
{
    const int wave = threadIdx.x >> 5;
    const int lane = threadIdx.x & 31;
    const int row  = blockIdx.x * 8 + wave;
    const float* p = X + (size_t)row * DIM;
    float s = 0.f;
#pragma unroll
    for (int t = 0; t < 8; ++t) {
        float v = p[lane + 32 * t];
        s = fmaf(v, v, s);
    }
#pragma unroll
    for (int m = 16; m >= 1; m >>= 1) s += __shfl_xor(s, m, 32);
    if (lane == 0) {
        sq[row] = s;
        ap[row] = 0u;           // 0.0f
        an[row] = 0x7F800000u;  // +inf
    }
}

// ---------------------------------------------------------------------------
// Kernel 2: fused gram-tile + distance + hard mining, async double-buffered.
// grid.x = 32 row-blocks (128 rows, 16 per wave)
// grid.y = 16 col-chunks (256 cols = 8 tiles of 32 cols; each tile's K=256
//          staged as two K-halves of 128 -> 16 pipeline steps)
// B tiles streamed into LDS with GLOBAL_LOAD_ASYNC_TO_LDS_B128 (ASYNCcnt),
// overlapped with the 2 interleaved fp32 WMMA accumulator chains.
// ---------------------------------------------------------------------------
#define LDSS      132                    // padded row stride (floats), 132%64==4
#define BUFFLOATS (32 * LDSS)            // one buffer: 32 cols x 128 K (+pad)

__launch_bounds__(256, 1)
__global__ void tl_mine_tiles(const float* __restrict__ X,
                              const int* __restrict__ labels,
                              const float* __restrict__ sq,
                              unsigned* __restrict__ ap,
                              unsigned* __restrict__ an)
{
    __shared__ float lds_b[2 * BUFFLOATS];   // ~33.8 KB double buffer

    const int tid  = threadIdx.x;
    const int wave = tid >> 5;
    const int lane = tid & 31;
    const int hi   = lane >> 4;              // K-half select (ISA A/B layout)
    const int l16  = lane & 15;
    const int r0   = blockIdx.x * 128 + wave * 16;
    const int cbase = blockIdx.y * 256;      // first column of this chunk

    // A fragment in registers: lane holds X[r0+l16][kt*4 + hi*2 + {0,1}]
    v2f a[64];
    {
        const float* apX = X + (size_t)(r0 + l16) * DIM + hi * 2;
#pragma unroll
        for (int kt = 0; kt < 64; ++kt)
            a[kt] = *(const v2f*)(apX + kt * 4);
    }

    // this wave's row metadata (C layout: VGPR r <-> row r + 8*hi)
    float sq_m[8]; int lb_m[8];
#pragma unroll
    for (int r = 0; r < 8; ++r) {
        int row = r0 + r + 8 * hi;
        sq_m[r] = sq[row];
        lb_m[r] = labels[row];
    }

    float maxap[8], minan[8];
#pragma unroll
    for (int r = 0; r < 8; ++r) { maxap[r] = 0.f; minan[r] = __builtin_inff(); }

    // issue one pipeline step's async fill: 32 cols x 128 K -> buffer step&1
    // each thread issues 4 b128 async DMAs (1024 float4 total)
    auto issue_fill = [&](int step) {
        const int c0 = cbase + (step >> 1) * 32;       // column base
        const int kh = (step & 1) << 7;                // K-half base (0 or 128)
        const float* gsrc = X + (size_t)c0 * DIM + kh;
        unsigned lbase = (unsigned)(uintptr_t)&lds_b[(step & 1) * BUFFLOATS];
#pragma unroll
        for (int i = 0; i < 4; ++i) {
            int f    = i * 256 + tid;                  // float4 index 0..1023
            int rrow = f >> 5;                         // 32 float4 per row
            int col4 = f & 31;
            unsigned    ldst = lbase + (unsigned)((rrow * LDSS + col4 * 4) * 4);
            const void* ga   = (const void*)(gsrc + (size_t)rrow * DIM + col4 * 4);
            asm volatile("global_load_async_to_lds_b128 %0, %1, off"
                         :: "v"(ldst), "v"(ga) : "memory");
        }
    };

    issue_fill(0);                                     // pipeline prologue

    for (int jp = 0; jp < 8; ++jp) {
        v8f acc0 = {0.f, 0.f, 0.f, 0.f, 0.f, 0.f, 0.f, 0.f};
        v8f acc1 = {0.f, 0.f, 0.f, 0.f, 0.f, 0.f, 0.f, 0.f};
#pragma unroll
        for (int s = 0; s < 2; ++s) {                  // K-halves; buffer = s
            asm volatile("s_wait_asynccnt 0" ::: "memory");
            __syncthreads();                           // fills visible to WGP
            if (jp * 2 + s + 1 < 16)                   // overlap next fill
                issue_fill(jp * 2 + s + 1);

            const float* bbase = &lds_b[s * BUFFLOATS];
            const float* bp0   = bbase + l16 * LDSS + hi * 2;
            const float* bp1   = bp0 + 16 * LDSS;
#pragma unroll
            for (int kt = 0; kt < 32; ++kt) {
                v2f b0 = *(const v2f*)(bp0 + kt * 4);
                v2f b1 = *(const v2f*)(bp1 + kt * 4);
                acc0 = __builtin_amdgcn_wmma_f32_16x16x4_f32(
                           false, a[s * 32 + kt], false, b0, (short)0, acc0, false, false);
                acc1 = __builtin_amdgcn_wmma_f32_16x16x4_f32(
                           false, a[s * 32 + kt], false, b1, (short)0, acc1, false, false);
            }
        }

        // fused distance + hard mining epilogue (registers + L2 only, no LDS)
        const int c0 = cbase + jp * 32;
        const float sq_n0 = sq[c0 + l16];
        const float sq_n1 = sq[c0 + 16 + l16];
        const int   lb_n0 = labels[c0 + l16];
        const int   lb_n1 = labels[c0 + 16 + l16];
#pragma unroll
        for (int r = 0; r < 8; ++r) {
            float d20 = fmaf(-2.f, acc0[r], sq_m[r] + sq_n0);
            float d21 = fmaf(-2.f, acc1[r], sq_m[r] + sq_n1);
            float d0  = sqrtf(fmaxf(d20, 1e-12f));
            float d1  = sqrtf(fmaxf(d21, 1e-12f));
            bool s0 = (lb_m[r] == lb_n0);
            bool s1 = (lb_m[r] == lb_n1);
            maxap[r] = s0 ? fmaxf(maxap[r], d0) : maxap[r];
            minan[r] = s0 ? minan[r] : fminf(minan[r], d0);
            maxap[r] = s1 ? fmaxf(maxap[r], d1) : maxap[r];
            minan[r] = s1 ? minan[r] : fminf(minan[r], d1);
        }
    }

    // reduce across the 16 lanes of each half (N direction)
#pragma unroll
    for (int m = 8; m >= 1; m >>= 1) {
#pragma unroll
        for (int r = 0; r < 8; ++r) {
            maxap[r] = fmaxf(maxap[r], __shfl_xor(maxap[r], m, 32));
            minan[r] = fminf(minan[r], __shfl_xor(minan[r], m, 32));
        }
    }
    // deterministic (order-independent) merge across col-chunks:
    // bit-pattern compare == float compare for nonneg finite values
    if (l16 == 0) {
#pragma unroll
        for (int r = 0; r < 8; ++r) {
            int row = r0 + r + 8 * hi;
            atomicMax(&ap[row], __float_as_uint(maxap[r]));
            atomicMin(&an[row], __float_as_uint(minan[r]));
        }
    }
}

// ---------------------------------------------------------------------------
// Kernel 3: single-block deterministic reduction -> mean margin loss
// ---------------------------------------------------------------------------
__global__ void tl_finalize(const unsigned* __restrict__ ap,
                            const unsigned* __restrict__ an,
                            float* __restrict__ out)
{
    const int tid = threadIdx.x;
    float s = 0.f;
    for (int row = tid; row < N_ROWS; row += 256) {
        float dap = __uint_as_float(ap[row]);
        float dan = __uint_as_float(an[row]);
        s += fmaxf(dap - dan + MARGIN, 0.f);
    }
#pragma unroll
    for (int m = 16; m >= 1; m >>= 1) s += __shfl_xor(s, m, 32);
    __shared__ float wsum[8];
    if ((tid & 31) == 0) wsum[tid >> 5] = s;
    __syncthreads();
    if (tid == 0) {
        float t = 0.f;
#pragma unroll
        for (int w = 0; w < 8; ++w) t += wsum[w];
        out[0] = t * (1.0f / N_ROWS);
    }
}

// ---------------------------------------------------------------------------
extern "C" void kernel_launch(void* const* d_in, const int* in_sizes, int n_in,
                              void* d_out, int out_size, void* d_ws, size_t ws_size,
                              hipStream_t stream)
{
    const float* X      = (const float*)d_in[0];
    const int*   labels = (const int*)d_in[1];   // per harness: integer -> const int*
    float*       out    = (float*)d_out;

    float*    sq = (float*)d_ws;                             // 4096 f32
    unsigned* ap = (unsigned*)((char*)d_ws + N_ROWS * 4);    // 4096 u32
    unsigned* an = ap + N_ROWS;                              // 4096 u32

    tl_rownorm_init<<<N_ROWS / 8, 256, 0, stream>>>(X, sq, ap, an);
    dim3 grid(32, 16);
    tl_mine_tiles<<<grid, 256, 0, stream>>>(X, labels, sq, ap, an);
    tl_finalize<<<1, 256, 0, stream>>>(ap, an, out);
}